// TAGConvolution_43731357008201
// MI455X (gfx1250) — compile-verified
//
#include <hip/hip_runtime.h>

// TAGConv: K=3 hops SpMM + concat + dense GEMM, fp32.
// N=50000, E=800000, F=64, OUT=64, concat width = 256.

typedef __attribute__((ext_vector_type(2))) float v2f;
typedef __attribute__((ext_vector_type(8))) float v8f;

#define FEAT 64
#define CONCAT 256

// ---------------- zero workspace ----------------
__global__ void tag_zero_kernel(float* __restrict__ p, long long count) {
    long long i = (long long)blockIdx.x * blockDim.x + threadIdx.x;
    if (i < count) p[i] = 0.0f;
}

// ---------------- SpMM scatter: dst[row] += val * src[col] ----------------
// one thread per (edge, feature); a wave covers 32 contiguous features of
// one edge (edge scalars are wave-uniform), gather of src[col*64+f] is
// contiguous, scatter via global_atomic_add_f32. h (12.8MB) is L2-resident.
__global__ __launch_bounds__(256) void tag_spmm_kernel(
    const int* __restrict__ erow, const int* __restrict__ ecol,
    const float* __restrict__ eval, const float* __restrict__ src,
    float* __restrict__ dst, int nedges)
{
    long long gid = (long long)blockIdx.x * blockDim.x + threadIdx.x;
    int e = (int)(gid >> 6);          // edge index
    int f = (int)(gid & 63);          // feature index
    if (e >= nedges) return;
    int r = erow[e];
    int c = ecol[e];
    float v = eval[e];
    float contrib = v * src[(long long)c * FEAT + f];
    atomicAdd(&dst[(long long)r * FEAT + f], contrib);
}

// ---------------- dense GEMM via V_WMMA_F32_16X16X4_F32 ----------------
// out[N,64] = [h0|h1|h2|h3][N,256] @ W[256,64] + b
// Each wave: one 16-row tile x all 64 cols (4 x 16-col tiles, 4 x v8f acc).
// A frag (16x4 f32, 2 VGPR): lanes 0-15 -> M=0..15, K={0,1}; lanes 16-31 -> K={2,3}
// B frag (4x16 f32, 2 VGPR): lanes 0-15 -> N=0..15, K={0,1}; lanes 16-31 -> K={2,3}
// C/D (16x16 f32, 8 VGPR): VGPR i: lanes 0-15 -> M=i, lanes 16-31 -> M=i+8; N=lane%16
//
// LDS holds W in K-pair-interleaved layout so each B fragment is ONE aligned
// ds_load_b64 into an even VGPR pair:
//   Wlds[(k>>1)*128 + n*2 + (k&1)] = W[k*64 + n]
__global__ __launch_bounds__(256) void tag_gemm_kernel(
    const float* __restrict__ h0, const float* __restrict__ h1,
    const float* __restrict__ h2, const float* __restrict__ h3,
    const float* __restrict__ W,  const float* __restrict__ bias,
    float* __restrict__ out, int n)
{
    __shared__ float Wlds[CONCAT * FEAT];   // 64 KB of 320 KB/WGP

    int tid = threadIdx.x;
    // cooperative W load into pair-interleaved layout (coalesced reads)
    for (int i = tid; i < CONCAT * FEAT; i += 256) {
        int k = i >> 6;       // 0..255
        int c = i & 63;       // 0..63
        Wlds[(k >> 1) * 128 + c * 2 + (k & 1)] = W[i];
    }
    __syncthreads();

    int wave = tid >> 5;                 // wave32 on gfx1250
    int lane = tid & 31;
    int rowTile = blockIdx.x * 8 + wave; // 16 rows per tile
    if (rowTile * 16 >= n) return;       // wave-uniform exit

    int laneM  = lane & 15;
    int khalf  = (lane >> 4) * 2;        // 0 for lanes 0-15, 2 for lanes 16-31
    int rowA   = rowTile * 16 + laneM;
    if (rowA >= n) rowA = n - 1;         // clamp loads; stores guarded below

    v8f acc0 = {}, acc1 = {}, acc2 = {}, acc3 = {};

    // per-lane LDS base: pair row ((k+khalf)>>1)*128 + laneM*2
    const float* ldsLane = &Wlds[(khalf >> 1) * 128 + laneM * 2];

    #pragma unroll
    for (int part = 0; part < 4; ++part) {
        const float* hb = (part == 0) ? h0 : (part == 1) ? h1 : (part == 2) ? h2 : h3;
        const float* arow = hb + (long long)rowA * FEAT;
        #pragma unroll
        for (int kk = 0; kk < FEAT; kk += 4) {
            int k = part * FEAT + kk;    // global K position, 0..255 (mult of 4)

            // A fragment: one aligned 8-byte global load
            v2f a = *(const v2f*)(arow + kk + khalf);

            // B fragments: one aligned 8-byte ds load each
            const float* bbase = ldsLane + (k >> 1) * 128;
            v2f b0 = *(const v2f*)(bbase);       // cols  0..15
            v2f b1 = *(const v2f*)(bbase + 32);  // cols 16..31
            v2f b2 = *(const v2f*)(bbase + 64);  // cols 32..47
            v2f b3 = *(const v2f*)(bbase + 96);  // cols 48..63

            acc0 = __builtin_amdgcn_wmma_f32_16x16x4_f32(false, a, false, b0, (short)0, acc0, false, false);
            acc1 = __builtin_amdgcn_wmma_f32_16x16x4_f32(false, a, false, b1, (short)0, acc1, false, false);
            acc2 = __builtin_amdgcn_wmma_f32_16x16x4_f32(false, a, false, b2, (short)0, acc2, false, false);
            acc3 = __builtin_amdgcn_wmma_f32_16x16x4_f32(false, a, false, b3, (short)0, acc3, false, false);
        }
    }

    // bias per lane, per col-tile
    float bv0 = bias[ 0 + laneM];
    float bv1 = bias[16 + laneM];
    float bv2 = bias[32 + laneM];
    float bv3 = bias[48 + laneM];

    int m0 = (lane < 16) ? 0 : 8;
    #pragma unroll
    for (int i = 0; i < 8; ++i) {
        int r = rowTile * 16 + m0 + i;
        if (r < n) {
            float* orow = out + (long long)r * FEAT;
            orow[ 0 + laneM] = acc0[i] + bv0;
            orow[16 + laneM] = acc1[i] + bv1;
            orow[32 + laneM] = acc2[i] + bv2;
            orow[48 + laneM] = acc3[i] + bv3;
        }
    }
}

extern "C" void kernel_launch(void* const* d_in, const int* in_sizes, int n_in,
                              void* d_out, int out_size, void* d_ws, size_t ws_size,
                              hipStream_t stream) {
    const float* x    = (const float*)d_in[0];
    const int*   erow = (const int*)  d_in[1];
    const int*   ecol = (const int*)  d_in[2];
    const float* eval = (const float*)d_in[3];
    const float* W    = (const float*)d_in[4];
    const float* bias = (const float*)d_in[5];
    float* out = (float*)d_out;

    int n      = in_sizes[0] / FEAT;   // N = 50000
    int nedges = in_sizes[1];          // E = 800000

    float* h1 = (float*)d_ws;
    float* h2 = h1 + (size_t)n * FEAT;
    float* h3 = h2 + (size_t)n * FEAT;

    // zero h1..h3 (contiguous)
    long long zcount = 3LL * n * FEAT;
    int zblocks = (int)((zcount + 255) / 256);
    tag_zero_kernel<<<zblocks, 256, 0, stream>>>(h1, zcount);

    // 3 SpMM hops
    long long ework = (long long)nedges * FEAT;
    int eblocks = (int)((ework + 255) / 256);
    tag_spmm_kernel<<<eblocks, 256, 0, stream>>>(erow, ecol, eval, x,  h1, nedges);
    tag_spmm_kernel<<<eblocks, 256, 0, stream>>>(erow, ecol, eval, h1, h2, nedges);
    tag_spmm_kernel<<<eblocks, 256, 0, stream>>>(erow, ecol, eval, h2, h3, nedges);

    // WMMA GEMM: 8 row-tiles (16 rows each) per block -> 128 rows/block
    int rowTiles = (n + 15) / 16;
    int gblocks  = (rowTiles + 7) / 8;
    tag_gemm_kernel<<<gblocks, 256, 0, stream>>>(x, h1, h2, h3, W, bias, out, n);
}